// construct_ABC_3178275799347
// MI455X (gfx1250) — compile-verified
//
#include <hip/hip_runtime.h>

typedef __attribute__((ext_vector_type(2))) float v2f;
typedef __attribute__((ext_vector_type(8))) float v8f;

#define BATCH  8
#define NPTS   4096
#define TILES  (NPTS / 16)
#define FHUGE  3.0e38f   // finite, keeps packed double out of NaN/Inf range

// Pack a non-negative finite float key + 32-bit index into a positive double.
// IEEE ordering of the doubles == (key, index) lexicographic ordering.
__device__ __forceinline__ double pack_di(float v, int idx) {
    return __hiloint2double(__float_as_int(v), idx);
}
__device__ __forceinline__ int unpack_idx(double d) {
    return __double2loint(d);
}

// One wave32 per 16-query row tile; scan 4096 candidates in 16-wide tiles.
// d2 = |p|^2 + |q|^2 - 2 p.q via one V_WMMA_F32_16X16X4_F32 per tile:
//   A row (16x4) = (-2px, -2py, -2pz, 1), B col (4x16) = (qx, qy, qz, |q|^2),
//   C initialized with |p_row|^2. Candidate loads are software-pipelined
//   one tile ahead to overlap L2 latency with WMMA + f64 top-2 reduction.
__global__ __launch_bounds__(32) void knn2_wmma_kernel(
    const float* __restrict__ coords,   // (8, 4096, 3)
    float* __restrict__ outA,           // (8, 4096, 3)
    float* __restrict__ outC,           // (8, 4096, 3)
    float* __restrict__ outI1,          // (8, 4096) as float
    float* __restrict__ outI2)          // (8, 4096) as float
{
    const int lane = threadIdx.x;              // 0..31
    const int wid  = blockIdx.x;               // 0..2047
    const int b    = wid / TILES;
    const int rowTile = wid % TILES;
    const int rowBase = rowTile * 16;
    const float* __restrict__ cb = coords + (size_t)b * NPTS * 3;

    const int half = (lane >= 16) ? 1 : 0;     // K-pair / row-half selector
    const int l16  = lane & 15;

    // ---- A operand (loop invariant): query rows of this tile ----
    const int qr = rowBase + l16;
    const float px = cb[qr * 3 + 0];
    const float py = cb[qr * 3 + 1];
    const float pz = cb[qr * 3 + 2];
    const float psq = px * px + py * py + pz * pz;
    v2f amat;
    amat[0] = half ? (-2.0f * pz) : (-2.0f * px);   // K=2 : K=0
    amat[1] = half ? 1.0f         : (-2.0f * py);   // K=3 : K=1

    // |p_row|^2 for the 8 rows this lane's accumulator elements cover
    // (element m -> row m + 8*half; that row's psq lives in lane m + 8*half).
    v8f crow;
#pragma unroll
    for (int m = 0; m < 8; ++m)
        crow[m] = __shfl(psq, m + 8 * half, 32);

    // ---- per-lane running top-2 as packed (d2, idx) doubles ----
    double s1[8], s2[8];
#pragma unroll
    for (int m = 0; m < 8; ++m) { s1[m] = pack_di(FHUGE, 0); s2[m] = pack_di(FHUGE, 0); }

    // ---- software-pipelined candidate scan: prefetch tile t+1 while computing t ----
    float nx = cb[l16 * 3 + 0];
    float ny = cb[l16 * 3 + 1];
    float nz = cb[l16 * 3 + 2];

#pragma unroll 4
    for (int t = 0; t < TILES; ++t) {
        const float qx = nx, qy = ny, qz = nz;   // tile t (already loaded)
        const int cand  = t * 16 + l16;          // this lane's candidate column
        const int cnext = (cand + 16) & (NPTS - 1);  // wrap keeps last prefetch in-bounds
        nx = cb[cnext * 3 + 0];                  // issue tile t+1 load now,
        ny = cb[cnext * 3 + 1];                  // consumed next iteration
        nz = cb[cnext * 3 + 2];

        const float qsq = qx * qx + qy * qy + qz * qz;
        v2f bmat;
        bmat[0] = half ? qz  : qx;              // K=2 : K=0
        bmat[1] = half ? qsq : qy;              // K=3 : K=1

        v8f acc = __builtin_amdgcn_wmma_f32_16x16x4_f32(
            /*neg_a=*/false, amat, /*neg_b=*/false, bmat,
            /*c_mod=*/(short)0, /*C=*/crow, /*reuse_a=*/false, /*reuse_b=*/false);

        if (t == rowTile) {                     // wave-uniform branch: diagonal tile
#pragma unroll
            for (int m = 0; m < 8; ++m) {
                const bool self = (l16 == m + 8 * half);   // cand == global row
                acc[m] = self ? FHUGE : acc[m];
            }
        }

#pragma unroll
        for (int m = 0; m < 8; ++m) {
            const float  v = fmaxf(acc[m], 0.0f);          // true d2 >= 0
            const double p = pack_di(v, cand);
            const double lose = fmax(p, s1[m]);            // loser of 1st place
            s1[m] = fmin(p, s1[m]);
            s2[m] = fmin(lose, s2[m]);
        }
    }

    // ---- cross-lane top-2 merge within each half-wave (16 lanes per row) ----
#pragma unroll
    for (int m = 0; m < 8; ++m) {
#pragma unroll
        for (int mask = 1; mask < 16; mask <<= 1) {
            const double o1 = __shfl_xor(s1[m], mask, 32);
            const double o2 = __shfl_xor(s2[m], mask, 32);
            const double hi = fmax(s1[m], o1);
            s1[m] = fmin(s1[m], o1);
            s2[m] = fmin(fmin(s2[m], o2), hi);
        }
    }

    // ---- write: lane r (0..7) of each half writes row rowBase + r + 8*half ----
    if (l16 < 8) {
        double d1 = s1[0], d2v = s2[0];
#pragma unroll
        for (int m = 1; m < 8; ++m) {
            const bool sel = (l16 == m);
            d1  = sel ? s1[m] : d1;
            d2v = sel ? s2[m] : d2v;
        }
        const int j1 = unpack_idx(d1);
        const int j2 = unpack_idx(d2v);
        const int row = rowBase + l16 + half * 8;
        const size_t rowG = (size_t)b * NPTS + row;
#pragma unroll
        for (int k = 0; k < 3; ++k) {
            outA[rowG * 3 + k] = cb[j1 * 3 + k];
            outC[rowG * 3 + k] = cb[j2 * 3 + k];
        }
        outI1[rowG] = (float)j1;
        outI2[rowG] = (float)j2;
    }
}

extern "C" void kernel_launch(void* const* d_in, const int* in_sizes, int n_in,
                              void* d_out, int out_size, void* d_ws, size_t ws_size,
                              hipStream_t stream) {
    (void)in_sizes; (void)n_in; (void)d_ws; (void)ws_size; (void)out_size;
    const float* coords = (const float*)d_in[0];
    // d_in[1] is the all-ones mask; the reference ignores it.

    float* out = (float*)d_out;
    float* outA  = out;                                   // 8*4096*3
    float* outC  = out + (size_t)BATCH * NPTS * 3;        // 8*4096*3
    float* outI1 = out + (size_t)2 * BATCH * NPTS * 3;    // 8*4096
    float* outI2 = outI1 + (size_t)BATCH * NPTS;          // 8*4096

    const int nWaves = BATCH * TILES;                     // 2048 row tiles
    knn2_wmma_kernel<<<nWaves, 32, 0, stream>>>(coords, outA, outC, outI1, outI2);
}